// Transducer_46102178955736
// MI455X (gfx1250) — compile-verified
//
#include <hip/hip_runtime.h>

#define NEGV (-1e30f)

constexpr int Bn = 128;   // batch
constexpr int Tn = 3000;  // time
constexpr int Cn = 64;    // classes (blank = 63)
constexpr int Ln = 128;   // target length
constexpr int Sn = 2 * Ln + 1;  // 257 extended states
constexpr int CHUNK = 16;       // timesteps staged per LDS chunk (WMMA M dim)
constexpr int XPAD = 68;        // xrow row stride (4 mod 64 -> conflict-free A reads)
constexpr int EPAD = 273;       // em row stride (8*273 % 64 != 0 -> D writes spread)

typedef float v2f __attribute__((ext_vector_type(2)));
typedef float v8f __attribute__((ext_vector_type(8)));

__global__ void ctc_zero_out(float* out) {
  if (threadIdx.x == 0) out[0] = 0.0f;
}

__global__ __launch_bounds__(256)
void ctc_forward(const float* __restrict__ x,   // [B, T, C] raw logits
                 const int* __restrict__ tgt,   // [B, L]
                 float* __restrict__ out) {     // scalar
  __shared__ float xrow[CHUNK * XPAD];   // staged logit rows
  __shared__ float em[CHUNK * EPAD];     // emissions for chunk: em[t][s]
  __shared__ float abuf[2][Sn + 3];      // double-buffered alpha, idx s+2 (0,1 = NEG pads)
  __shared__ int   zsh[Sn];              // extended label sequence
  __shared__ float dvals[CHUNK];         // log-softmax denominators per row

  const int b    = blockIdx.x;
  const int tid  = threadIdx.x;
  const int lane = tid & 31;
  const int wave = tid >> 5;
  const float* xb = x + (size_t)b * Tn * Cn;

  // z: blanks (63) at even s, targets at odd s
  for (int s = tid; s < Sn; s += 256)
    zsh[s] = (s & 1) ? tgt[b * Ln + (s >> 1)] : (Cn - 1);
  if (tid < 2) { abuf[0][tid] = NEGV; abuf[1][tid] = NEGV; }
  __syncthreads();

  // ---- loop-invariant one-hot B fragments for the emission matmul.
  // Wave `wave` owns s-tiles [wave*32, wave*32+16) and [wave*32+16, wave*32+32).
  // ISA B layout (16x16x4, f32): N = lane%16; VGPR0 holds K = khalf, VGPR1 K = khalf+1,
  // with khalf = 0 (lanes 0-15) / 2 (lanes 16-31), mirroring the A-matrix table.
  const int khalf = (lane >= 16) ? 2 : 0;
  const int mrow  = lane & 15;            // A: M (= timestep) = lane % 16
  const int scol  = wave * 32 + (lane & 15);
#if __has_builtin(__builtin_amdgcn_wmma_f32_16x16x4_f32)
  v2f bf0[16], bf1[16];
  {
    const int zs0 = zsh[scol];
    const int zs1 = zsh[scol + 16];
    #pragma unroll
    for (int k = 0; k < 16; ++k) {
      const int c0 = k * 4 + khalf;
      bf0[k].x = (zs0 == c0) ? 1.0f : 0.0f;
      bf0[k].y = (zs0 == c0 + 1) ? 1.0f : 0.0f;
      bf1[k].x = (zs1 == c0) ? 1.0f : 0.0f;
      bf1[k].y = (zs1 == c0 + 1) ? 1.0f : 0.0f;
    }
  }
#endif

  int cur = 0;  // abuf[cur] = alpha_{t-1}
  for (int t0 = 0; t0 < Tn; t0 += CHUNK) {
    const int tcount = (Tn - t0 < CHUNK) ? (Tn - t0) : CHUNK;

    // ---- phase 1: async-DMA 16 rows (16x64 f32 = 4KB) straight into LDS.
    // GVS mode: SGPR base = row block start, per-lane 32-bit byte offset,
    // per-lane LDS dest address (generic shared ptr low 32 bits = LDS offset).
    {
      const int idx = tid * 4;
      const int r = idx >> 6;          // row in chunk
      const int c = idx & 63;          // column
      if (r < tcount) {
        const unsigned lds_addr = (unsigned)(uintptr_t)(&xrow[r * XPAD + c]);
        const unsigned voff = (unsigned)(tid * 16);  // bytes into the chunk
        const unsigned long long sbase =
            (unsigned long long)(uintptr_t)(xb + (size_t)t0 * Cn);
        asm volatile("global_load_async_to_lds_b128 %0, %1, %2"
                     :: "v"(lds_addr), "v"(voff), "s"(sbase)
                     : "memory");
      }
      if (t0 + CHUNK + r < Tn)  // global_prefetch_b8: warm L2 for next chunk
        __builtin_prefetch(xb + (size_t)(t0 + CHUNK + r) * Cn + c, 0, 1);
      asm volatile("s_wait_asynccnt 0x0" ::: "memory");
    }
    __syncthreads();

    // ---- phase 2: d[t] = logsumexp over the 64 classes (wave32 shuffle reduce)
    for (int r = wave * 2; r < wave * 2 + 2; ++r) {
      const float x0 = xrow[r * XPAD + lane];
      const float x1 = xrow[r * XPAD + 32 + lane];
      float m = fmaxf(x0, x1);
      #pragma unroll
      for (int off = 16; off; off >>= 1) m = fmaxf(m, __shfl_xor(m, off, 32));
      float sm = __expf(x0 - m) + __expf(x1 - m);
      #pragma unroll
      for (int off = 16; off; off >>= 1) sm += __shfl_xor(sm, off, 32);
      if (lane == 0) dvals[r] = m + __logf(sm);
    }
    __syncthreads();

    // ---- phase 3: emissions em[t][s] = x[t][z[s]] - d[t] for 16 t x 256 s
    // Exact one-hot matmul on the f32 matrix core:
    //   Em(16x16) = sum_k A(16x4 slice of X) @ OneHot(4x16)   [V_WMMA_F32_16X16X4_F32]
#if __has_builtin(__builtin_amdgcn_wmma_f32_16x16x4_f32)
    {
      v8f acc0 = {}, acc1 = {};
      #pragma unroll
      for (int k = 0; k < 16; ++k) {
        const int c0 = k * 4 + khalf;
        v2f a; a.x = xrow[mrow * XPAD + c0]; a.y = xrow[mrow * XPAD + c0 + 1];
        acc0 = __builtin_amdgcn_wmma_f32_16x16x4_f32(
            false, a, false, bf0[k], (short)0, acc0, false, false);
        acc1 = __builtin_amdgcn_wmma_f32_16x16x4_f32(
            false, a, false, bf1[k], (short)0, acc1, false, false);
      }
      const int thi = (lane >= 16) ? 8 : 0;  // D: VGPR r -> M=r (lo half) / r+8 (hi)
      #pragma unroll
      for (int r = 0; r < 8; ++r) {
        const int t = r + thi;
        em[t * EPAD + scol]      = acc0[r] - dvals[t];
        em[t * EPAD + scol + 16] = acc1[r] - dvals[t];
      }
    }
#else
    for (int i = tid; i < CHUNK * 256; i += 256) {
      const int t = i >> 8, s = i & 255;
      em[t * EPAD + s] = xrow[t * XPAD + zsh[s]] - dvals[t];
    }
#endif
    if (tid < CHUNK)  // s = 256 (trailing blank) handled by gather
      em[tid * EPAD + 256] = xrow[tid * XPAD + (Cn - 1)] - dvals[tid];
    __syncthreads();

    // ---- phase 4: sequential alpha recursion over the chunk (1 barrier / step)
    for (int tt = 0; tt < tcount; ++tt) {
      const float* ap = abuf[cur];
      float* an = abuf[cur ^ 1];
      const int tg = t0 + tt;
      for (int s = tid; s < Sn; s += 256) {
        const float e = em[tt * EPAD + s];
        float nv;
        if (tg == 0) {
          nv = (s < 2) ? e : NEGV;
        } else {
          const float a0 = ap[s + 2];
          const float a1 = ap[s + 1];
          const float a2 = (s & 1) ? ap[s] : NEGV;  // skip only into non-blank states
          const float m = fmaxf(fmaxf(a0, a1), a2);
          const float r3 = __expf(a0 - m) + __expf(a1 - m) + __expf(a2 - m);
          nv = e + m + __logf(r3);
        }
        an[s + 2] = nv;
      }
      __syncthreads();
      cur ^= 1;
    }
  }

  // ---- final: ll = LSE(alpha_T[S-2], alpha_T[S-1]); accumulate mean(-ll)
  if (tid == 0) {
    const float aS2 = abuf[cur][Sn];      // state S-2 at idx (S-2)+2
    const float aS1 = abuf[cur][Sn + 1];  // state S-1 at idx (S-1)+2
    const float m = fmaxf(aS2, aS1);
    const float ll = m + __logf(__expf(aS2 - m) + __expf(aS1 - m));
    atomicAdd(out, -ll * (1.0f / Bn));
  }
}

extern "C" void kernel_launch(void* const* d_in, const int* in_sizes, int n_in,
                              void* d_out, int out_size, void* d_ws, size_t ws_size,
                              hipStream_t stream) {
  (void)in_sizes; (void)n_in; (void)out_size; (void)d_ws; (void)ws_size;
  const float* x  = (const float*)d_in[0];
  const int* tgt  = (const int*)d_in[1];
  float* out      = (float*)d_out;
  ctc_zero_out<<<1, 32, 0, stream>>>(out);
  ctc_forward<<<Bn, 256, 0, stream>>>(x, tgt, out);
}